// SpatialEmbedding_69441031242282
// MI455X (gfx1250) — compile-verified
//
#include <hip/hip_runtime.h>
#include <math.h>

typedef float v2f __attribute__((ext_vector_type(2)));
typedef float v8f __attribute__((ext_vector_type(8)));

#define BATCH   4
#define NTOK    1024
#define KFREQ   256
#define D_MODEL 512
#define ITILE   16
#define JCHUNK  128
#define NCHUNK  (NTOK / JCHUNK)
#define INV4PI  0.07957747154594767f   // 1/(4*pi)

// Tensor Data Mover path (device pass only; host pass / other toolchains fall back).
#if defined(__AMDGCN__) && __has_builtin(__builtin_amdgcn_tensor_load_to_lds) && \
    __has_builtin(__builtin_amdgcn_s_wait_tensorcnt)
#define USE_TDM 1
typedef unsigned int v4u __attribute__((ext_vector_type(4)));
typedef int          v4i __attribute__((ext_vector_type(4)));
typedef int          v8i __attribute__((ext_vector_type(8)));
#endif

// Helmholtz Green's-function positional embedding.
// One workgroup = (batch b, 16 query rows). 256 threads = 8 wave32s.
//  * coords[b] staged into LDS by the TDM (tensor_load_to_lds), norms precomputed.
//  * 16x16 squared-distance tiles via V_WMMA_F32_16X16X4_F32:
//      sq = |xi|^2 + |xj|^2 - 2 xi.xj  (A=(-2x,-2y,-2z,1), B=(x,y,z,|xj|^2), C=|xi|^2)
//  * (r, inv=mask/(4*pi*r)) tiles double-buffered in LDS; produce(c+1) overlaps consume(c).
//  * Each thread owns one frequency t; v_sin/v_cos take revolutions, and
//    k_t/(2*pi) = 1/wl_t, so the phase is simply r * inv_wl (|phase| < 256 revs).
__global__ __launch_bounds__(256) void helmholtz_embed_kernel(
    const float* __restrict__ coords,           // [B, N, 3] f32
    const unsigned char* __restrict__ pad_mask, // [B, N] bool (True = padded)
    float* __restrict__ out)                    // [B, N, 512] f32
{
    __shared__ float4 sc[NTOK];        // x, y, z, |x|^2
    __shared__ float  svalidf[NTOK];   // 1.0 = valid token, 0.0 = padded
    __shared__ __align__(16) union {
        float  raw[NTOK * 3];                   // TDM staging (dead after prologue)
        float2 pair[2][ITILE][JCHUNK];          // ping-pong (r, inv) tiles
    } sbuf;

    const int tid  = threadIdx.x;      // 0..255 == frequency index
    const int lane = tid & 31;
    const int wave = tid >> 5;         // 0..7
    const int b    = blockIdx.x >> 6;
    const int i0   = (blockIdx.x & 63) * ITILE;
    const int half = lane >> 4;        // 0: lanes 0-15, 1: lanes 16-31
    const int l15  = lane & 15;

    // ---------------- prologue: stage coords + norms + validity ----------------
#ifdef USE_TDM
    if (tid < 32) {                    // wave 0 issues the tensor DMA
        unsigned long long ga = (unsigned long long)(const void*)(coords + (size_t)b * NTOK * 3);
        unsigned int lds_off  = (unsigned int)(uintptr_t)(void*)sbuf.raw;
        // D# group 0: count=1, lds_addr, global_addr[56:0], type=2
        v4u g0 = { 1u, lds_off,
                   (unsigned int)(ga & 0xFFFFFFFFu),
                   (unsigned int)((ga >> 32) & 0x01FFFFFFu) | (2u << 30) };
        // D# group 1: data_size=4B; 1-row tile of 3072 elements; stride0=3072
        v8i g1 = { 0x20000,                 // data_size=2 -> 4 bytes
                   (int)(3072u << 16),      // tensor_dim0[15:0]
                   (int)(1u << 16),         // tensor_dim0 hi=0 | tensor_dim1=1
                   (int)(3072u << 16),      // tensor_dim1 hi=0 | tile_dim0=3072
                   1,                       // tile_dim1=1, tile_dim2=0
                   3072,                    // tensor_dim0_stride
                   0, 0 };
        v4i gz = { 0, 0, 0, 0 };
#if defined(__clang_major__) && (__clang_major__ >= 23)
        v8i gz8 = { 0, 0, 0, 0, 0, 0, 0, 0 };
        __builtin_amdgcn_tensor_load_to_lds(g0, g1, gz, gz, gz8, 0);
#else
        __builtin_amdgcn_tensor_load_to_lds(g0, g1, gz, gz, 0);
#endif
        __builtin_amdgcn_s_wait_tensorcnt(0);
    }
    __syncthreads();
    const float4* gq = reinterpret_cast<const float4*>(sbuf.raw);
#else
    const float4* gq = reinterpret_cast<const float4*>(coords + (size_t)b * NTOK * 3);
#endif
    {
        float4 q0 = gq[tid * 3 + 0];
        float4 q1 = gq[tid * 3 + 1];
        float4 q2 = gq[tid * 3 + 2];
        const int j0 = tid * 4;
        sc[j0 + 0] = make_float4(q0.x, q0.y, q0.z, q0.x*q0.x + q0.y*q0.y + q0.z*q0.z);
        sc[j0 + 1] = make_float4(q0.w, q1.x, q1.y, q0.w*q0.w + q1.x*q1.x + q1.y*q1.y);
        sc[j0 + 2] = make_float4(q1.z, q1.w, q2.x, q1.z*q1.z + q1.w*q1.w + q2.x*q2.x);
        sc[j0 + 3] = make_float4(q2.y, q2.z, q2.w, q2.y*q2.y + q2.z*q2.z + q2.w*q2.w);
        uchar4 mv = *reinterpret_cast<const uchar4*>(pad_mask + (size_t)b * NTOK + j0);
        svalidf[j0 + 0] = mv.x ? 0.0f : 1.0f;
        svalidf[j0 + 1] = mv.y ? 0.0f : 1.0f;
        svalidf[j0 + 2] = mv.z ? 0.0f : 1.0f;
        svalidf[j0 + 3] = mv.w ? 0.0f : 1.0f;
    }
    __syncthreads();

    // ---- per-thread wavenumber: wl_t = min + (max-min)*(base^t - 1)/(base-1) ----
    float tt     = (float)tid * (1.0f / (KFREQ - 1));
    float wl     = 3.7f + (20.0f - 3.7f) * (exp2f(tt * 4.3219280948873626f) - 1.0f) * (1.0f / 19.0f);
    float inv_wl = 1.0f / wl;          // = k_t / (2*pi)

    // ---- hoisted WMMA A/C operands + row validity (i-tile invariant) ----
    float4 ci = sc[i0 + l15];          // A row M = l15
    v2f amat;                          // lanes 0-15: K=0,1 ; lanes 16-31: K=2,3
    amat[0] = half ? (-2.0f * ci.z) : (-2.0f * ci.x);
    amat[1] = half ? 1.0f            : (-2.0f * ci.y);
    v8f cmat;                          // VGPR v holds row M = v + 8*half
    float vi[8];
    #pragma unroll
    for (int v = 0; v < 8; ++v) {
        cmat[v] = sc[i0 + v + half * 8].w;
        vi[v]   = svalidf[i0 + v + half * 8];
    }

    float acc_re[ITILE], acc_im[ITILE];
    #pragma unroll
    for (int i = 0; i < ITILE; ++i) { acc_re[i] = 0.0f; acc_im[i] = 0.0f; }

    // ---- produce one (r, inv) tile: wave w owns 16 columns, branchless mask ----
    auto produce = [&](int c, int buf) {
        const int jg = c * JCHUNK + wave * 16 + l15;   // this lane's column
        float4 cj = sc[jg];
        v2f bmat;                      // lanes 0-15: rows K=0,2 ; lanes 16-31: K=1,3
        bmat[0] = half ? cj.y : cj.x;
        bmat[1] = half ? cj.w : cj.z;
        v8f d = __builtin_amdgcn_wmma_f32_16x16x4_f32(
            false, amat, false, bmat, (short)0, cmat, false, false);
        float vj = svalidf[jg];
        #pragma unroll
        for (int v = 0; v < 8; ++v) {
            int irow = v + half * 8;
            float sq  = d[v];
            // |sq| instead of max(sq,0): negative sq is only rounding noise near the
            // (masked) diagonal, and fabs folds into v_sqrt's input modifier for free.
            float r   = __builtin_amdgcn_sqrtf(__builtin_fabsf(sq));
            float m   = (i0 + irow == jg) ? 0.0f : vi[v] * vj;   // mask diag + padding
            float inv = m * (INV4PI * __builtin_amdgcn_rcpf(fmaxf(r, 1e-20f)));
            sbuf.pair[buf][irow][wave * 16 + l15] = make_float2(r, inv);
        }
    };

    // ---- consume a tile: every thread applies its frequency to 16x128 pairs ----
    auto consume = [&](int buf) {
        #pragma unroll
        for (int i = 0; i < ITILE; ++i) {
            const float4* row = reinterpret_cast<const float4*>(&sbuf.pair[buf][i][0]);
            float re = acc_re[i], im = acc_im[i];
            #pragma unroll 4
            for (int q = 0; q < JCHUNK / 2; ++q) {
                float4 p = row[q];                      // (r0, inv0, r1, inv1)
                float ph0 = p.x * inv_wl;
                re = fmaf(__builtin_amdgcn_cosf(ph0), p.y, re);
                im = fmaf(__builtin_amdgcn_sinf(ph0), p.y, im);
                float ph1 = p.z * inv_wl;
                re = fmaf(__builtin_amdgcn_cosf(ph1), p.w, re);
                im = fmaf(__builtin_amdgcn_sinf(ph1), p.w, im);
            }
            acc_re[i] = re; acc_im[i] = im;
        }
    };

    // ---- software-pipelined chunk loop: one barrier per chunk ----
    produce(0, 0);
    for (int c = 0; c < NCHUNK; ++c) {
        __syncthreads();               // buf[c&1] ready; buf[(c+1)&1] fully consumed
        if (c + 1 < NCHUNK) produce(c + 1, (c + 1) & 1);
        consume(c & 1);
    }

    // ---- out[b, i0+i, 2t] = re, out[b, i0+i, 2t+1] = im (coalesced b64 stores) ----
    float2* o2 = reinterpret_cast<float2*>(out);
    #pragma unroll
    for (int i = 0; i < ITILE; ++i) {
        o2[((size_t)b * NTOK + i0 + i) * (D_MODEL / 2) + tid] =
            make_float2(acc_re[i], acc_im[i]);
    }
}

extern "C" void kernel_launch(void* const* d_in, const int* in_sizes, int n_in,
                              void* d_out, int out_size, void* d_ws, size_t ws_size,
                              hipStream_t stream) {
    (void)in_sizes; (void)n_in; (void)d_ws; (void)ws_size; (void)out_size;
    const float*         coords = (const float*)d_in[0];
    const unsigned char* mask   = (const unsigned char*)d_in[1];
    float*               out    = (float*)d_out;

    dim3 grid(BATCH * (NTOK / ITILE));   // 256 workgroups
    dim3 block(256);                     // 8 wave32s
    helmholtz_embed_kernel<<<grid, block, 0, stream>>>(coords, mask, out);
}